// RoutingStrategy_78546361909750
// MI455X (gfx1250) — compile-verified
//
#include <hip/hip_runtime.h>
#include <hip/hip_bf16.h>
#include <stdint.h>

#define Bh 4
#define Th 1024
#define Dh 1024
#define Eh 8
#define FFh 2048
#define BT (Bh*Th)
#define NASSIGN (BT*2)
#define MAXROWS (NASSIGN + Eh*16)
#define MAXTILES (MAXROWS/16)

typedef float v2f __attribute__((ext_vector_type(2)));
typedef float v8f __attribute__((ext_vector_type(8)));

// ---------------------------------------------------------------- init
__global__ void moe_init(int* counts, int* cursors, int* perm_token,
                         int* row_to_assign, int* tile_e) {
    int i = blockIdx.x * blockDim.x + threadIdx.x;
    int stride = gridDim.x * blockDim.x;
    if (i < Eh) { counts[i] = 0; cursors[i] = 0; }
    for (int r = i; r < MAXROWS; r += stride) { perm_token[r] = 0; row_to_assign[r] = -1; }
    for (int t = i; t < MAXTILES; t += stride) tile_e[t] = -1;
}

// ---------------------------------------------------------------- router
__global__ void moe_router(const float* __restrict__ x, const float* __restrict__ rw,
                           const float* __restrict__ rb, const float* __restrict__ temp,
                           float* __restrict__ scores) {
    int i = blockIdx.x * blockDim.x + threadIdx.x;       // over BT*E
    if (i >= BT * Eh) return;
    int t = i / Eh, e = i % Eh;
    const float4* xr = (const float4*)(x + (size_t)t * Dh);
    const float4* wr = (const float4*)(rw + (size_t)e * Dh);
    float acc = 0.f;
    for (int j = 0; j < Dh / 4; ++j) {
        float4 a = xr[j], b = wr[j];
        acc += a.x * b.x + a.y * b.y + a.z * b.z + a.w * b.w;
    }
    float tc = fmaxf(temp[0], 0.1f);
    scores[i] = (acc + rb[e]) / tc;
}

// ---------------------------------------------------------------- sinkhorn + top2
__global__ __launch_bounds__(1024)
void moe_sinkhorn_topk(const float* __restrict__ scores, float* __restrict__ topk_g,
                       int* __restrict__ topk_e, int* __restrict__ counts) {
    int b = blockIdx.x;
    int t = threadIdx.x;                  // token in batch, 0..1023
    int lane = t & 31, wid = t >> 5;      // wave32
    const float* row = scores + ((size_t)b * Th + t) * Eh;
    float la[Eh];
#pragma unroll
    for (int e = 0; e < Eh; ++e) la[e] = row[e];

    __shared__ float red[Eh][33];
    __shared__ float colv[Eh];

    for (int it = 0; it < 3; ++it) {
        // row logsumexp (over E, per thread)
        float m = la[0];
#pragma unroll
        for (int e = 1; e < Eh; ++e) m = fmaxf(m, la[e]);
        float s = 0.f;
#pragma unroll
        for (int e = 0; e < Eh; ++e) s += expf(la[e] - m);
        float lse = m + logf(s);
#pragma unroll
        for (int e = 0; e < Eh; ++e) la[e] -= lse;

        // column max over T
#pragma unroll
        for (int e = 0; e < Eh; ++e) {
            float v = la[e];
            for (int off = 16; off > 0; off >>= 1) v = fmaxf(v, __shfl_down(v, off));
            if (lane == 0) red[e][wid] = v;
        }
        __syncthreads();
        if (t < Eh) {
            float mm = -3.0e38f;
            for (int w = 0; w < 32; ++w) mm = fmaxf(mm, red[t][w]);
            colv[t] = mm;
        }
        __syncthreads();
        float cm[Eh];
#pragma unroll
        for (int e = 0; e < Eh; ++e) cm[e] = colv[e];
        __syncthreads();
        // column sum of exp
#pragma unroll
        for (int e = 0; e < Eh; ++e) {
            float v = expf(la[e] - cm[e]);
            for (int off = 16; off > 0; off >>= 1) v += __shfl_down(v, off);
            if (lane == 0) red[e][wid] = v;
        }
        __syncthreads();
        if (t < Eh) {
            float ss = 0.f;
            for (int w = 0; w < 32; ++w) ss += red[t][w];
            colv[t] = ss;
        }
        __syncthreads();
#pragma unroll
        for (int e = 0; e < Eh; ++e) la[e] -= cm[e] + logf(colv[e]);
        __syncthreads();
    }

    // gates + row normalize
    float g[Eh]; float rs = 0.f;
#pragma unroll
    for (int e = 0; e < Eh; ++e) { g[e] = expf(la[e]); rs += g[e]; }
    rs += 1e-8f;
#pragma unroll
    for (int e = 0; e < Eh; ++e) g[e] /= rs;

    // top-2 (first occurrence on ties, like lax.top_k)
    int e0 = 0; float v0 = g[0];
#pragma unroll
    for (int e = 1; e < Eh; ++e) if (g[e] > v0) { v0 = g[e]; e0 = e; }
    int e1 = -1; float v1 = -3.0e38f;
#pragma unroll
    for (int e = 0; e < Eh; ++e) if (e != e0 && g[e] > v1) { v1 = g[e]; e1 = e; }
    float gs = v0 + v1 + 1e-8f;

    size_t aid = ((size_t)b * Th + t) * 2;
    topk_e[aid] = e0;     topk_g[aid] = v0 / gs;
    topk_e[aid + 1] = e1; topk_g[aid + 1] = v1 / gs;
    atomicAdd(&counts[e0], 1);
    atomicAdd(&counts[e1], 1);
}

// ---------------------------------------------------------------- offsets + tile table
__global__ void moe_offsets(const int* __restrict__ counts, int* offs, int* cursors,
                            int* tile_e, int* tile_row, int* ntiles) {
    if (threadIdx.x != 0 || blockIdx.x != 0) return;
    int total = 0, tile = 0;
    for (int e = 0; e < Eh; ++e) {
        offs[e] = total;
        cursors[e] = 0;
        int rows = counts[e];
        int padded = (rows + 15) & ~15;
        for (int i = 0; i < padded / 16; ++i) {
            tile_e[tile] = e;
            tile_row[tile] = total + i * 16;
            ++tile;
        }
        total += padded;
    }
    ntiles[0] = tile;
}

// ---------------------------------------------------------------- scatter assignments
__global__ void moe_scatter(const int* __restrict__ topk_e, const int* __restrict__ offs,
                            int* cursors, int* perm_token, int* row_to_assign) {
    int aid = blockIdx.x * blockDim.x + threadIdx.x;
    if (aid >= NASSIGN) return;
    int t = aid >> 1;
    int e = topk_e[aid];
    int slot = atomicAdd(&cursors[e], 1);
    int r = offs[e] + slot;
    perm_token[r] = t;
    row_to_assign[r] = aid;
}

// ---------------------------------------------------------------- expert FFN with fp32 WMMA
// One block = one tile of 16 tokens of one expert, 16 waves (512 threads).
// Stage 1: h[16,FF] = relu(X*wg)*(X*wp)   (each wave owns 128 FF cols)
// Stage 2: y[16,D]  = h*wo                (each wave owns 64 D cols)
__global__ __launch_bounds__(512)
void moe_ffn_wmma(const float* __restrict__ x,
                  const float* __restrict__ wg,
                  const float* __restrict__ wp,
                  const float* __restrict__ wo,
                  const int* __restrict__ perm_token,
                  const int* __restrict__ row_to_assign,
                  const int* __restrict__ tile_e,
                  const int* __restrict__ tile_row,
                  float* __restrict__ contrib) {
    __shared__ float Xs[16 * Dh];    // 64 KB
    __shared__ float Hs[16 * FFh];   // 128 KB  (320 KB/WGP LDS)

    int e = tile_e[blockIdx.x];
    if (e < 0) return;               // uniform: padding tile
    int rowbase = tile_row[blockIdx.x];

    int tid = threadIdx.x;
    int lane = tid & 31, wave = tid >> 5;   // 16 waves
    int n = lane & 15, half = lane >> 4;    // WMMA lane decomposition

    // Stage X rows into LDS via CDNA5 async global->LDS DMA (no VGPR round trip).
    // Wave w stages permuted token row w: 1024 f32 = 8 x (32 lanes x b128).
    {
        int tok = perm_token[rowbase + wave];
        unsigned goff  = (unsigned)(tok * Dh + lane * 4) * 4u;              // byte offset vs x base
        unsigned ldsa  = (unsigned)(uintptr_t)(Xs + wave * Dh + lane * 4);  // LDS byte address (low 32b of flat)
#pragma unroll
        for (int i = 0; i < 8; ++i) {
            asm volatile("global_load_async_to_lds_b128 %0, %1, %2"
                         :: "v"(ldsa + (unsigned)i * 512u),
                            "v"(goff + (unsigned)i * 512u),
                            "s"(x)
                         : "memory");
        }
#if __has_builtin(__builtin_amdgcn_s_wait_asynccnt)
        __builtin_amdgcn_s_wait_asynccnt(0);
#else
        asm volatile("s_wait_asynccnt 0x0" ::: "memory");
#endif
    }
    __syncthreads();

    const float* wgE = wg + (size_t)e * Dh * FFh;
    const float* wpE = wp + (size_t)e * Dh * FFh;
    const float* woE = wo + (size_t)e * FFh * Dh;

    // ---- Stage 1 ----
    {
        v8f accG[8] = {};
        v8f accP[8] = {};
        int fbase = wave * 128;
        const float* xrow = Xs + n * Dh;     // A: M = lane&15
        for (int d0 = 0; d0 < Dh; d0 += 4) {
            // A tile 16x4 fp32: lanes<16 hold K={d0,d0+1}, lanes>=16 hold K={d0+2,d0+3}
            v2f a = *(const v2f*)(xrow + d0 + 2 * half);
            int dB = d0 + 2 * half;
#pragma unroll
            for (int ft = 0; ft < 8; ++ft) {
                int f = fbase + ft * 16 + n;  // B: N = lane&15
                v2f bg, bp;
                bg.x = wgE[(size_t)dB * FFh + f];
                bg.y = wgE[(size_t)(dB + 1) * FFh + f];
                bp.x = wpE[(size_t)dB * FFh + f];
                bp.y = wpE[(size_t)(dB + 1) * FFh + f];
                accG[ft] = __builtin_amdgcn_wmma_f32_16x16x4_f32(
                    false, a, false, bg, (short)0, accG[ft], false, false);
                accP[ft] = __builtin_amdgcn_wmma_f32_16x16x4_f32(
                    false, a, false, bp, (short)0, accP[ft], false, false);
            }
        }
        // h = relu(g) * p -> LDS
#pragma unroll
        for (int ft = 0; ft < 8; ++ft) {
#pragma unroll
            for (int i = 0; i < 8; ++i) {
                int M = i + 8 * half;        // C/D layout: VGPR i holds rows i, i+8
                int f = fbase + ft * 16 + n;
                float gv = accG[ft][i];
                gv = gv > 0.f ? gv : 0.f;
                Hs[M * FFh + f] = gv * accP[ft][i];
            }
        }
    }
    __syncthreads();

    // ---- Stage 2 ----
    {
        v8f acc[4] = {};
        int dbase = wave * 64;
        const float* hrow = Hs + n * FFh;
        for (int f0 = 0; f0 < FFh; f0 += 4) {
            v2f a = *(const v2f*)(hrow + f0 + 2 * half);
            int fB = f0 + 2 * half;
#pragma unroll
            for (int dt = 0; dt < 4; ++dt) {
                int c = dbase + dt * 16 + n;
                v2f bv;
                bv.x = woE[(size_t)fB * Dh + c];
                bv.y = woE[(size_t)(fB + 1) * Dh + c];
                acc[dt] = __builtin_amdgcn_wmma_f32_16x16x4_f32(
                    false, a, false, bv, (short)0, acc[dt], false, false);
            }
        }
        // epilogue: scatter valid rows to per-assignment contrib buffer
#pragma unroll
        for (int i = 0; i < 8; ++i) {
            int M = i + 8 * half;
            int aid = row_to_assign[rowbase + M];
            if (aid >= 0) {
                float* dst = contrib + (size_t)aid * Dh + dbase;
#pragma unroll
                for (int dt = 0; dt < 4; ++dt) dst[dt * 16 + n] = acc[dt][i];
            }
        }
    }
}

// ---------------------------------------------------------------- gather + residual
__global__ void moe_gather(const float* __restrict__ x, const float* __restrict__ contrib,
                           const float* __restrict__ topk_g, float* __restrict__ out) {
    int i = blockIdx.x * blockDim.x + threadIdx.x;   // over BT*D/4
    if (i >= BT * Dh / 4) return;
    int t = i / (Dh / 4);
    int c = i % (Dh / 4);
    const float4* x4 = (const float4*)x;
    const float4* c4 = (const float4*)contrib;
    float4 xv = x4[i];
    float g0 = topk_g[t * 2], g1 = topk_g[t * 2 + 1];
    float4 a = c4[(size_t)(t * 2) * (Dh / 4) + c];
    float4 b = c4[(size_t)(t * 2 + 1) * (Dh / 4) + c];
    float4 o;
    o.x = xv.x + g0 * a.x + g1 * b.x;
    o.y = xv.y + g0 * a.y + g1 * b.y;
    o.z = xv.z + g0 * a.z + g1 * b.z;
    o.w = xv.w + g0 * a.w + g1 * b.w;
    ((float4*)out)[i] = o;
}

// ---------------------------------------------------------------- launch
extern "C" void kernel_launch(void* const* d_in, const int* in_sizes, int n_in,
                              void* d_out, int out_size, void* d_ws, size_t ws_size,
                              hipStream_t stream) {
    (void)in_sizes; (void)n_in; (void)out_size; (void)ws_size;
    const float* x    = (const float*)d_in[0];
    const float* rw   = (const float*)d_in[1];
    const float* rb   = (const float*)d_in[2];
    const float* temp = (const float*)d_in[3];
    const float* wg   = (const float*)d_in[4];
    const float* wp   = (const float*)d_in[5];
    const float* wo   = (const float*)d_in[6];
    float* out = (float*)d_out;

    char* ws = (char*)d_ws;
    size_t off = 0;
    auto alloc = [&](size_t bytes) -> void* {
        void* p = ws + off;
        off = (off + bytes + 255) & ~(size_t)255;
        return p;
    };
    float* contrib      = (float*)alloc((size_t)NASSIGN * Dh * sizeof(float)); // 33.5 MB
    float* scores       = (float*)alloc((size_t)BT * Eh * sizeof(float));
    float* topk_g       = (float*)alloc((size_t)NASSIGN * sizeof(float));
    int*   topk_e       = (int*)  alloc((size_t)NASSIGN * sizeof(int));
    int*   counts       = (int*)  alloc(Eh * sizeof(int));
    int*   offs         = (int*)  alloc(Eh * sizeof(int));
    int*   cursors      = (int*)  alloc(Eh * sizeof(int));
    int*   perm_token   = (int*)  alloc(MAXROWS * sizeof(int));
    int*   row_to_assign= (int*)  alloc(MAXROWS * sizeof(int));
    int*   tile_e       = (int*)  alloc(MAXTILES * sizeof(int));
    int*   tile_row     = (int*)  alloc(MAXTILES * sizeof(int));
    int*   ntiles       = (int*)  alloc(sizeof(int));

    moe_init<<<64, 256, 0, stream>>>(counts, cursors, perm_token, row_to_assign, tile_e);
    moe_router<<<(BT * Eh + 255) / 256, 256, 0, stream>>>(x, rw, rb, temp, scores);
    moe_sinkhorn_topk<<<Bh, 1024, 0, stream>>>(scores, topk_g, topk_e, counts);
    moe_offsets<<<1, 1, 0, stream>>>(counts, offs, cursors, tile_e, tile_row, ntiles);
    moe_scatter<<<NASSIGN / 256, 256, 0, stream>>>(topk_e, offs, cursors, perm_token, row_to_assign);
    moe_ffn_wmma<<<MAXTILES, 512, 0, stream>>>(x, wg, wp, wo, perm_token, row_to_assign,
                                               tile_e, tile_row, contrib);
    moe_gather<<<(BT * Dh / 4 + 255) / 256, 256, 0, stream>>>(x, contrib, topk_g, out);
}